// Net_69904887710304
// MI455X (gfx1250) — compile-verified
//
#include <hip/hip_runtime.h>

#define NTHREADS 256
#define PITCH     72   // LDS row pitch in floats (= staged region width, 16B-chunk multiple)
#define RMAX      70   // max staged rows (64 + 2*3)

typedef float v2f __attribute__((ext_vector_type(2)));

// ---- CDNA5 async global->LDS copies (ASYNCcnt path) ----------------------
__device__ __forceinline__ void async_b128(unsigned lds_byte, unsigned goff,
                                           unsigned long long base) {
  asm volatile("global_load_async_to_lds_b128 %0, %1, %2"
               :: "v"(lds_byte), "v"(goff), "s"(base) : "memory");
}
__device__ __forceinline__ void async_b32(unsigned lds_byte, unsigned goff,
                                          unsigned long long base) {
  asm volatile("global_load_async_to_lds_b32 %0, %1, %2"
               :: "v"(lds_byte), "v"(goff), "s"(base) : "memory");
}
__device__ __forceinline__ void wait_async0() {
  asm volatile("s_wait_asynccnt 0" ::: "memory");
}

// ---- fused (conv3x3 x K) + maxpool2x2 stage ------------------------------
// LDS tile: PITCH(=72) cols x R(=64+2K) rows.
//   LDS col 0  <-> global col bx*64 - 4   (16B aligned; left/right halo = 4 >= K)
//   LDS row 0  <-> global row by*64 - K
// Conv pass j computes rows [j+1, R-1-j), cols [5-K+j, 67+K-j)  (funnel to the
// 64x64 pre-pool tile at rows [K, K+64), cols [4, 68)). Each lane computes a
// column PAIR per row with packed fp32 FMAs (v_pk_fma_f32 path).
template <int K>
__global__ __launch_bounds__(NTHREADS)
void stage_conv_pool(const float* __restrict__ in, int Win,
                     const float* __restrict__ wp0,
                     const float* __restrict__ wp1,
                     const float* __restrict__ wp2,
                     float* __restrict__ out) {
  __shared__ float bufA[RMAX * PITCH];
  __shared__ float bufB[RMAX * PITCH];

  constexpr int R      = 64 + 2 * K;       // staged rows
  constexpr int CPR    = PITCH / 4;        // 18 b128 chunks per row
  constexpr int NCHUNK = R * CPR;

  const int tid = (int)threadIdx.x;
  const int tx  = tid & 31;                // lane
  const int ty  = tid >> 5;                // wave id 0..7

  const int gx0 = (int)blockIdx.x * 64 - 4;   // global col of LDS col 0 (aligned)
  const int gy0 = (int)blockIdx.y * 64 - K;   // global row of LDS row 0

  // uniform weight loads (scalar path)
  float w[K][9];
#pragma unroll
  for (int i = 0; i < 9; ++i) w[0][i] = wp0[i];
#pragma unroll
  for (int i = 0; i < 9; ++i) w[1][i] = wp1[i];
  if constexpr (K == 3) {
#pragma unroll
    for (int i = 0; i < 9; ++i) w[2][i] = wp2[i];
  }

  // ---- stage halo'd tile into LDS: B128 async interior, B32/zero edges ---
  const unsigned long long base = (unsigned long long)(uintptr_t)in;
  for (int q = tid; q < NCHUNK; q += NTHREADS) {
    const int r  = q / CPR;
    const int cg = q - r * CPR;
    const int c  = cg * 4;
    const int gr = gy0 + r;
    const int gc = gx0 + c;
    const unsigned lds_byte = (unsigned)(uintptr_t)&bufA[r * PITCH + c];
    if (gr >= 0 && gr < Win && gc >= 0 && gc + 3 < Win) {
      async_b128(lds_byte, (unsigned)(gr * Win + gc) * 4u, base);
    } else {
#pragma unroll
      for (int e = 0; e < 4; ++e) {
        const int gce = gc + e;
        if (gr >= 0 && gr < Win && gce >= 0 && gce < Win) {
          async_b32(lds_byte + 4u * e, (unsigned)(gr * Win + gce) * 4u, base);
        } else {
          bufA[r * PITCH + c + e] = 0.0f;   // SAME zero padding
        }
      }
    }
  }
  wait_async0();
  __syncthreads();

  // ---- K conv passes, ping-pong in LDS, packed column-pair rolling window
  float* src = bufA;
  float* dst = bufB;
#pragma unroll
  for (int j = 0; j < K; ++j) {
    const int lo_r = j + 1;
    const int hi_r = R - 1 - j;
    const int lo_c = 5 - K + j;
    const int hi_c = 67 + K - j;
    const int NP   = (hi_c - lo_c) >> 1;   // column pairs (width is even)
    const int ROWS = hi_r - lo_r;
    const int CH   = (ROWS + 7) / 8;       // rows per wave-row strip
    const int r0   = lo_r + ty * CH;
    const int r1   = (r0 + CH < hi_r) ? (r0 + CH) : hi_r;

    const v2f W0 = {w[j][0], w[j][0]}, W1 = {w[j][1], w[j][1]}, W2 = {w[j][2], w[j][2]};
    const v2f W3 = {w[j][3], w[j][3]}, W4 = {w[j][4], w[j][4]}, W5 = {w[j][5], w[j][5]};
    const v2f W6 = {w[j][6], w[j][6]}, W7 = {w[j][7], w[j][7]}, W8 = {w[j][8], w[j][8]};

#pragma unroll
    for (int cc = 0; cc < 2; ++cc) {
      const int pi = tx + 32 * cc;         // pair index
      const int c  = lo_c + 2 * pi;        // left column of the pair
      if (pi < NP && r0 < hi_r) {
        const float* p = &src[(r0 - 1) * PITCH + (c - 1)];
        float a0 = p[0], a1 = p[1], a2 = p[2], a3 = p[3];
        p += PITCH;
        float b0 = p[0], b1 = p[1], b2 = p[2], b3 = p[3];
        float* pd = &dst[r0 * PITCH + c];
        for (int r = r0; r < r1; ++r) {
          p += PITCH;
          const float e0 = p[0], e1 = p[1], e2 = p[2], e3 = p[3];
          v2f acc = ((v2f){a0, a1}) * W0;
          acc = __builtin_elementwise_fma((v2f){a1, a2}, W1, acc);
          acc = __builtin_elementwise_fma((v2f){a2, a3}, W2, acc);
          acc = __builtin_elementwise_fma((v2f){b0, b1}, W3, acc);
          acc = __builtin_elementwise_fma((v2f){b1, b2}, W4, acc);
          acc = __builtin_elementwise_fma((v2f){b2, b3}, W5, acc);
          acc = __builtin_elementwise_fma((v2f){e0, e1}, W6, acc);
          acc = __builtin_elementwise_fma((v2f){e1, e2}, W7, acc);
          acc = __builtin_elementwise_fma((v2f){e2, e3}, W8, acc);
          pd[0] = acc.x;
          pd[1] = acc.y;
          pd += PITCH;
          a0 = b0; a1 = b1; a2 = b2; a3 = b3;
          b0 = e0; b1 = e1; b2 = e2; b3 = e3;
        }
      }
    }
    __syncthreads();
    float* t = src; src = dst; dst = t;
  }

  // ---- 2x2 maxpool of rows [K, K+64), cols [4, 68); store 32x32 tile -----
  const int Wout = Win >> 1;
  const int ox0 = (int)blockIdx.x * 32;
  const int oy0 = (int)blockIdx.y * 32;
#pragma unroll
  for (int pr0 = 0; pr0 < 32; pr0 += 8) {
    const int pr = pr0 + ty;
    const int pc = tx;                         // 32 lanes = 32 cols
    const int sr = K + 2 * pr;
    const int sc = 4 + 2 * pc;
    const float a0 = src[sr * PITCH + sc];
    const float a1 = src[sr * PITCH + sc + 1];
    const float b0 = src[(sr + 1) * PITCH + sc];
    const float b1 = src[(sr + 1) * PITCH + sc + 1];
    const float m  = fmaxf(fmaxf(a0, a1), fmaxf(b0, b1));
    out[(size_t)(oy0 + pr) * Wout + (ox0 + pc)] = m;
  }
}

extern "C" void kernel_launch(void* const* d_in, const int* in_sizes, int n_in,
                              void* d_out, int out_size, void* d_ws, size_t ws_size,
                              hipStream_t stream) {
  (void)n_in; (void)out_size; (void)ws_size;

  const float* x = (const float*)d_in[0];
  const float* w[10];
  for (int i = 0; i < 10; ++i) w[i] = (const float*)d_in[1 + i];

  // H == W, derive from flat element count of x (power-of-two sized image)
  int W = 1;
  while ((long long)W * (long long)W < (long long)in_sizes[0]) W <<= 1;

  const size_t n1 = (size_t)(W / 2) * (W / 2);
  float* o1  = (float*)d_ws;            // stage1 out (W/2)^2
  float* o2  = o1 + n1;                 // stage2 out (W/4)^2
  float* o3  = (float*)d_ws;            // stage3 out (W/8)^2, aliases dead o1
  float* out = (float*)d_out;           // (W/16)^2

  dim3 blk(NTHREADS);

  dim3 g1(W / 64, W / 64);
  stage_conv_pool<2><<<g1, blk, 0, stream>>>(x, W, w[0], w[1], nullptr, o1);

  dim3 g2((W / 2) / 64, (W / 2) / 64);
  stage_conv_pool<2><<<g2, blk, 0, stream>>>(o1, W / 2, w[2], w[3], nullptr, o2);

  dim3 g3((W / 4) / 64, (W / 4) / 64);
  stage_conv_pool<3><<<g3, blk, 0, stream>>>(o2, W / 4, w[4], w[5], w[6], o3);

  dim3 g4((W / 8) / 64, (W / 8) / 64);
  stage_conv_pool<3><<<g4, blk, 0, stream>>>(o3, W / 8, w[7], w[8], w[9], out);
}